// GNNEncoder_85057532330087
// MI455X (gfx1250) — compile-verified
//
#include <hip/hip_runtime.h>
#include <hip/hip_bf16.h>

typedef __attribute__((ext_vector_type(2))) float v2f;
typedef __attribute__((ext_vector_type(8))) float v8f;

#define NN 50000
#define NE 800000
#define D  128
#define BN_EPS 1e-5f

// ---------------------------------------------------------------- utility
__global__ __launch_bounds__(256) void zero_kernel(float* __restrict__ p, int n) {
    int i = blockIdx.x * 256 + threadIdx.x;
    if (i < n) p[i] = 0.0f;
}

// ---------------------------------------------------------------- degree count
__global__ __launch_bounds__(256) void count_kernel(const int* __restrict__ ei,
                                                    float* __restrict__ cnt) {
    int e = blockIdx.x * 256 + threadIdx.x;
    if (e < NE) {
        int d = ei[NE + e];   // dst row of edge_index
        __hip_atomic_fetch_add(&cnt[d], 1.0f, __ATOMIC_RELAXED, __HIP_MEMORY_SCOPE_AGENT);
    }
}

__global__ __launch_bounds__(256) void inv_kernel(const float* __restrict__ cnt,
                                                  float* __restrict__ inv) {
    int i = blockIdx.x * 256 + threadIdx.x;
    if (i < NN) inv[i] = 1.0f / fmaxf(cnt[i], 1.0f);
}

// ---------------------------------------------------------------- edge scatter-add
// one thread handles a float4 chunk of one edge's message: 32 threads / edge
__global__ __launch_bounds__(256) void scatter_kernel(const float* __restrict__ h,
                                                      const int* __restrict__ ei,
                                                      float* __restrict__ agg) {
    int t = blockIdx.x * 256 + threadIdx.x;
    int e = t >> 5;
    if (e >= NE) return;
    int c = (t & 31) << 2;
    int s = ei[e];
    int d = ei[NE + e];
    const float4 v = *(const float4*)(h + (size_t)s * D + c);
    float* p = agg + (size_t)d * D + c;
    __hip_atomic_fetch_add(p + 0, v.x, __ATOMIC_RELAXED, __HIP_MEMORY_SCOPE_AGENT);
    __hip_atomic_fetch_add(p + 1, v.y, __ATOMIC_RELAXED, __HIP_MEMORY_SCOPE_AGENT);
    __hip_atomic_fetch_add(p + 2, v.z, __ATOMIC_RELAXED, __HIP_MEMORY_SCOPE_AGENT);
    __hip_atomic_fetch_add(p + 3, v.w, __ATOMIC_RELAXED, __HIP_MEMORY_SCOPE_AGENT);
}

// ---------------------------------------------------------------- fused SAGE GEMM
// out[n,:] = (agg[n,:]*inv[n]) @ Wl^T + bl + h[n,:] @ Wr^T   (+ optional ReLU)
// block = 256 threads = 8 wave32; block owns 16 rows; wave w owns N-columns [16w,16w+16)
template <int RELU>
__global__ __launch_bounds__(256) void sage_gemm_kernel(const float* __restrict__ agg,
                                                        const float* __restrict__ inv,
                                                        const float* __restrict__ h,
                                                        const float* __restrict__ Wl,
                                                        const float* __restrict__ bl,
                                                        const float* __restrict__ Wr,
                                                        float* __restrict__ out) {
    __shared__ float sWl[D * D];    // 64 KB
    __shared__ float sWr[D * D];    // 64 KB
    __shared__ float sA[16 * D];    // mean rows, 8 KB
    __shared__ float sH[16 * D];    // self rows, 8 KB

    const int tid  = threadIdx.x;
    const int row0 = blockIdx.x * 16;

    // stage weights (float4) into LDS
    {
        const float4* wl4 = (const float4*)Wl;
        const float4* wr4 = (const float4*)Wr;
        float4* swl4 = (float4*)sWl;
        float4* swr4 = (float4*)sWr;
        #pragma unroll 4
        for (int i = tid; i < (D * D) / 4; i += 256) {
            swl4[i] = wl4[i];
            swr4[i] = wr4[i];
        }
        // stage the 16 mean rows (agg*inv) and 16 self rows
        float4* sa4 = (float4*)sA;
        float4* sh4 = (float4*)sH;
        for (int i = tid; i < (16 * D) / 4; i += 256) {   // 512 float4
            int r = i >> 5;                // row within tile (32 float4 per row)
            int g = row0 + r;
            float4 a = ((const float4*)agg)[(size_t)g * (D / 4) + (i & 31)];
            float  s = inv[g];
            a.x *= s; a.y *= s; a.z *= s; a.w *= s;
            sa4[i] = a;
            sh4[i] = ((const float4*)h)[(size_t)g * (D / 4) + (i & 31)];
        }
    }
    __syncthreads();

    const int lane  = tid & 31;
    const int wave  = tid >> 5;            // N tile 0..7
    const int mrow  = lane & 15;           // A-row within tile
    const int khalf = (lane >> 4) << 1;    // 0 or 2 (K sub-offset per ISA layout)
    const int col   = wave * 16 + (lane & 15);   // output column 0..127

    v8f acc = {};
    #pragma unroll
    for (int k = 0; k < D; k += 4) {
        const int kk = k + khalf;
        v2f a1 = *(const v2f*)(sA  + mrow * D + kk);
        v2f b1 = *(const v2f*)(sWl + col  * D + kk);   // B[k][n] = Wl[n][k]
        acc = __builtin_amdgcn_wmma_f32_16x16x4_f32(false, a1, false, b1,
                                                    (short)0, acc, false, false);
        v2f a2 = *(const v2f*)(sH  + mrow * D + kk);
        v2f b2 = *(const v2f*)(sWr + col  * D + kk);
        acc = __builtin_amdgcn_wmma_f32_16x16x4_f32(false, a2, false, b2,
                                                    (short)0, acc, false, false);
    }

    // C/D layout: VGPR j holds M = j + 8*(lane>=16), N = lane%16 (per N-tile)
    const float bias  = bl[col];
    const int   mbase = (lane >> 4) << 3;
    #pragma unroll
    for (int j = 0; j < 8; ++j) {
        float v = acc[j] + bias;
        if (RELU) v = fmaxf(v, 0.0f);
        out[(size_t)(row0 + mbase + j) * D + col] = v;
    }
}

// ---------------------------------------------------------------- batch-norm stats
__global__ __launch_bounds__(256) void bn_stats_kernel(const float* __restrict__ h,
                                                       float* __restrict__ mu,
                                                       float* __restrict__ var) {
    __shared__ float ssum[256];
    __shared__ float ssq[256];
    const int f = blockIdx.x;             // feature 0..127
    float s = 0.0f, q = 0.0f;
    for (int r = threadIdx.x; r < NN; r += 256) {
        float v = h[(size_t)r * D + f];
        s += v;
        q += v * v;
    }
    ssum[threadIdx.x] = s;
    ssq[threadIdx.x]  = q;
    __syncthreads();
    for (int w = 128; w > 0; w >>= 1) {
        if (threadIdx.x < w) {
            ssum[threadIdx.x] += ssum[threadIdx.x + w];
            ssq[threadIdx.x]  += ssq[threadIdx.x + w];
        }
        __syncthreads();
    }
    if (threadIdx.x == 0) {
        float m = ssum[0] / (float)NN;
        mu[f]  = m;
        var[f] = ssq[0] / (float)NN - m * m;   // population var, matches jnp.var
    }
}

__global__ __launch_bounds__(256) void bn_apply_kernel(float* __restrict__ h,
                                                       const float* __restrict__ mu,
                                                       const float* __restrict__ var,
                                                       const float* __restrict__ gamma,
                                                       const float* __restrict__ beta) {
    int i = blockIdx.x * 256 + threadIdx.x;
    if (i < NN * D) {
        int f = i & (D - 1);
        float v = (h[i] - mu[f]) * rsqrtf(var[f] + BN_EPS) * gamma[f] + beta[f];
        h[i] = fmaxf(v, 0.0f);
    }
}

// ---------------------------------------------------------------- launcher
extern "C" void kernel_launch(void* const* d_in, const int* in_sizes, int n_in,
                              void* d_out, int out_size, void* d_ws, size_t ws_size,
                              hipStream_t stream) {
    const float* x      = (const float*)d_in[0];
    const int*   ei     = (const int*)d_in[1];
    const float* Wl0    = (const float*)d_in[2];
    const float* bl0    = (const float*)d_in[3];
    const float* Wr0    = (const float*)d_in[4];
    const float* Wl1    = (const float*)d_in[5];
    const float* bl1    = (const float*)d_in[6];
    const float* Wr1    = (const float*)d_in[7];
    const float* Wl2    = (const float*)d_in[8];
    const float* bl2    = (const float*)d_in[9];
    const float* Wr2    = (const float*)d_in[10];
    const float* gamma1 = (const float*)d_in[11];
    const float* beta1  = (const float*)d_in[12];
    float* out = (float*)d_out;

    float* ws  = (float*)d_ws;
    float* agg = ws;                         // NN*D
    float* h0  = agg + (size_t)NN * D;       // NN*D
    float* h1  = h0  + (size_t)NN * D;       // NN*D
    float* cnt = h1  + (size_t)NN * D;       // NN
    float* inv = cnt + NN;                   // NN
    float* mu  = inv + NN;                   // D
    float* var = mu + D;                     // D

    const int ND = NN * D;                   // 6,400,000
    dim3 b256(256);
    const int ngrid = (NN + 255) / 256;
    const int zgrid = (ND + 255) / 256;
    const int egrid = (NE + 255) / 256;
    const int sgrid = (NE * 32 + 255) / 256; // 100,000 blocks
    const int ggrid = NN / 16;               // 3,125 blocks (exact)

    // edge degrees (shared by all layers)
    zero_kernel<<<ngrid, b256, 0, stream>>>(cnt, NN);
    count_kernel<<<egrid, b256, 0, stream>>>(ei, cnt);
    inv_kernel<<<ngrid, b256, 0, stream>>>(cnt, inv);

    // layer 0: h0 = relu(sage(x))
    zero_kernel<<<zgrid, b256, 0, stream>>>(agg, ND);
    scatter_kernel<<<sgrid, b256, 0, stream>>>(x, ei, agg);
    sage_gemm_kernel<1><<<ggrid, b256, 0, stream>>>(agg, inv, x, Wl0, bl0, Wr0, h0);

    // layer 1: h1 = sage(h0); h1 = relu(batch_norm(h1))
    zero_kernel<<<zgrid, b256, 0, stream>>>(agg, ND);
    scatter_kernel<<<sgrid, b256, 0, stream>>>(h0, ei, agg);
    sage_gemm_kernel<0><<<ggrid, b256, 0, stream>>>(agg, inv, h0, Wl1, bl1, Wr1, h1);
    bn_stats_kernel<<<D, b256, 0, stream>>>(h1, mu, var);
    bn_apply_kernel<<<zgrid, b256, 0, stream>>>(h1, mu, var, gamma1, beta1);

    // layer 2: out = sage(h1)
    zero_kernel<<<zgrid, b256, 0, stream>>>(agg, ND);
    scatter_kernel<<<sgrid, b256, 0, stream>>>(h1, ei, agg);
    sage_gemm_kernel<0><<<ggrid, b256, 0, stream>>>(agg, inv, h1, Wl2, bl2, Wr2, out);
}